// NLpObs_20203526160575
// MI455X (gfx1250) — compile-verified
//
#include <hip/hip_runtime.h>

typedef __attribute__((ext_vector_type(4))) float f4;
typedef __attribute__((ext_vector_type(2))) float v2f;
typedef __attribute__((ext_vector_type(8))) float v8f;

#define N_B              8
#define VEC_PER_BATCH    983040          // 15*512*512 / 4
#define BLOCKS_PER_BATCH 240
#define ITERS            16              // 240*16*256 == 983040, exact
#define THREADS          256
#define NWAVES           (THREADS / 32)

// acc += (finite(inp) ? (inp - x) : 0)^2 for 4 packed elements
__device__ __forceinline__ float sq_diff_acc(f4 xv, f4 iv, float acc) {
#pragma unroll
    for (int j = 0; j < 4; ++j) {
        float o = iv[j];
        unsigned u = __float_as_uint(o);
        // finite <=> exponent field != all-ones (rejects NaN and +/-Inf)
        bool fin = (u & 0x7F800000u) != 0x7F800000u;
        float d = fin ? (o - xv[j]) : 0.0f;
        acc = fmaf(d, d, acc);
    }
    return acc;
}

// Exact wave32 sum of `acc` across all lanes using v_wmma_f32_16x16x4_f32.
// A = ones(16x4), B = {acc lanes in b0, zeros in b1}. Every row of D equals
// the column-sum vector, so summing D row 0 (d[0], lanes 0..15) over N gives
// the total of all 32 accumulators regardless of B's internal element layout.
__device__ __forceinline__ float wave_sum_wmma(float acc) {
    v2f a;  a[0] = 1.0f; a[1] = 1.0f;
    v2f bm; bm[0] = acc; bm[1] = 0.0f;
    v8f c = {};
    v8f d = __builtin_amdgcn_wmma_f32_16x16x4_f32(
        /*neg_a=*/false, a, /*neg_b=*/false, bm,
        /*c_mod=*/(short)0, c, /*reuse_a=*/false, /*reuse_b=*/false);
    float w = d[0];                 // row M=0: N = lane (lanes 0..15)
    w += __shfl_xor(w, 8, 32);
    w += __shfl_xor(w, 4, 32);
    w += __shfl_xor(w, 2, 32);
    w += __shfl_xor(w, 1, 32);
    return w;                       // lane 0 holds the wave total
}

__global__ __launch_bounds__(THREADS)
void nlpobs_partial(const float* __restrict__ x,
                    const float* __restrict__ inp,
                    float* __restrict__ partials) {
    const int bx  = blockIdx.x;     // chunk within batch
    const int b   = blockIdx.y;     // batch
    const int tid = threadIdx.x;

    const f4* __restrict__ x4 = (const f4*)x;
    const f4* __restrict__ i4 = (const f4*)inp;
    const int base = b * VEC_PER_BATCH + bx * (ITERS * THREADS) + tid;

    float acc = 0.0f;
#pragma unroll 4
    for (int it = 0; it < ITERS; ++it) {
        const int idx = base + it * THREADS;
        // streamed once, footprint > L2 -> non-temporal policy
        f4 xv = __builtin_nontemporal_load(&x4[idx]);
        f4 iv = __builtin_nontemporal_load(&i4[idx]);
        acc = sq_diff_acc(xv, iv, acc);
    }

    // EXEC is all-ones here (uniform trip count, no divergence) -> WMMA legal
    float w = wave_sum_wmma(acc);

    __shared__ float smem[NWAVES];
    const int wave = tid >> 5;
    const int lane = tid & 31;
    if (lane == 0) smem[wave] = w;
    __syncthreads();

    if (wave == 0) {
        float s = (lane < NWAVES) ? smem[lane] : 0.0f;
        s += __shfl_xor(s, 4, 32);
        s += __shfl_xor(s, 2, 32);
        s += __shfl_xor(s, 1, 32);
        if (lane == 0) partials[b * BLOCKS_PER_BATCH + bx] = s;
    }
}

__global__ __launch_bounds__(THREADS)
void nlpobs_final(const float* __restrict__ partials,
                  float* __restrict__ out) {
    const int b   = blockIdx.x;
    const int tid = threadIdx.x;
    float v = (tid < BLOCKS_PER_BATCH) ? partials[b * BLOCKS_PER_BATCH + tid]
                                       : 0.0f;
    v += __shfl_xor(v, 16, 32);
    v += __shfl_xor(v, 8, 32);
    v += __shfl_xor(v, 4, 32);
    v += __shfl_xor(v, 2, 32);
    v += __shfl_xor(v, 1, 32);

    __shared__ float smem[NWAVES];
    const int wave = tid >> 5;
    const int lane = tid & 31;
    if (lane == 0) smem[wave] = v;
    __syncthreads();

    if (tid == 0) {
        float t = 0.0f;
#pragma unroll
        for (int i = 0; i < NWAVES; ++i) t += smem[i];
        out[b] = -1000.0f * t;      // -1/NOISE, NOISE = 0.001
    }
}

extern "C" void kernel_launch(void* const* d_in, const int* in_sizes, int n_in,
                              void* d_out, int out_size, void* d_ws, size_t ws_size,
                              hipStream_t stream) {
    (void)in_sizes; (void)n_in; (void)out_size; (void)ws_size;
    const float* x   = (const float*)d_in[0];   // "x"
    const float* inp = (const float*)d_in[1];   // "inp"
    float* out       = (float*)d_out;           // 8 floats
    float* partials  = (float*)d_ws;            // 8*240 floats scratch

    dim3 grid(BLOCKS_PER_BATCH, N_B);
    nlpobs_partial<<<grid, THREADS, 0, stream>>>(x, inp, partials);
    nlpobs_final<<<N_B, THREADS, 0, stream>>>(partials, out);
}